// GatedGraphConvolution_79809082295059
// MI455X (gfx1250) — compile-verified
//
#include <hip/hip_runtime.h>
#include <hip/hip_bf16.h>

// ---------------- problem constants (from reference setup_inputs) ------------
#define BDIM   4
#define TDIM   4
#define BT     16          // B*T
#define NNODES 65536
#define FIN    16
#define FOUT   32
#define SFEAT  8
#define SLAB   (BT * FIN)  // 256 floats = 1KB per node

typedef __attribute__((ext_vector_type(2))) float v2f;
typedef __attribute__((ext_vector_type(8))) float v8f;

// ---------------------------------------------------------------------------
// Kernel 1: gate = max(sig(static[:3]@Wg1+b1)*sig(static[3:]@Wg2+b2), min_gate)
//           x[n][bt][f] = water[bt][n][f] * gate ; agg zeroed.
// Thread id = n*16 + bt -> x/agg writes are fully coalesced (16 threads cover
// one contiguous 1KB slab).
// ---------------------------------------------------------------------------
__global__ void gate_scale_kernel(const float* __restrict__ water,
                                  const float* __restrict__ stat,
                                  const float* __restrict__ Wg1,
                                  const float* __restrict__ bg1,
                                  const float* __restrict__ Wg2,
                                  const float* __restrict__ bg2,
                                  const float* __restrict__ min_gate,
                                  float* __restrict__ x,
                                  float* __restrict__ agg) {
  int id = blockIdx.x * blockDim.x + threadIdx.x;   // [0, N*BT)
  int bt = id & (BT - 1);
  int n  = id >> 4;

  const float* s = stat + ((size_t)bt * NNODES + n) * SFEAT;
  float z1 = s[0] * Wg1[0] + s[1] * Wg1[1] + s[2] * Wg1[2] + bg1[0];
  float z2 = s[3] * Wg2[0] + s[4] * Wg2[1] + s[5] * Wg2[2] +
             s[6] * Wg2[3] + s[7] * Wg2[4] + bg2[0];
  float g = (1.0f / (1.0f + expf(-z1))) * (1.0f / (1.0f + expf(-z2)));
  g = fmaxf(g, min_gate[0]);

  const float4* w4 = (const float4*)(water + ((size_t)bt * NNODES + n) * FIN);
  float4* x4 = (float4*)(x   + (size_t)n * SLAB + bt * FIN);
  float4* a4 = (float4*)(agg + (size_t)n * SLAB + bt * FIN);
  float4 zero = make_float4(0.f, 0.f, 0.f, 0.f);
#pragma unroll
  for (int j = 0; j < 4; ++j) {
    float4 w = w4[j];
    w.x *= g; w.y *= g; w.z *= g; w.w *= g;
    x4[j] = w;
    a4[j] = zero;
  }
}

// ---------------------------------------------------------------------------
// Kernel 2: SpMM scatter. One wave (32 lanes) per edge; each lane moves 8 of
// the 256 floats in the node slab: agg[row] += val * x[col].
// x (64MB) is L2-resident on MI455X (192MB L2), atomics resolve in L2.
// ---------------------------------------------------------------------------
__global__ void spmm_kernel(const int*   __restrict__ rows,
                            const int*   __restrict__ cols,
                            const float* __restrict__ vals,
                            const float* __restrict__ x,
                            float* __restrict__ agg,
                            int E) {
  int gid  = blockIdx.x * blockDim.x + threadIdx.x;
  int eid  = gid >> 5;
  int lane = gid & 31;
  if (eid >= E) return;

  int   r = rows[eid];
  int   c = cols[eid];
  float v = vals[eid];

  const float4* xs = (const float4*)(x + (size_t)c * SLAB + lane * 8);
  float4 a = xs[0];
  float4 b = xs[1];
  float* dst = agg + (size_t)r * SLAB + lane * 8;
  atomicAdd(dst + 0, v * a.x);
  atomicAdd(dst + 1, v * a.y);
  atomicAdd(dst + 2, v * a.z);
  atomicAdd(dst + 3, v * a.w);
  atomicAdd(dst + 4, v * b.x);
  atomicAdd(dst + 5, v * b.y);
  atomicAdd(dst + 6, v * b.z);
  atomicAdd(dst + 7, v * b.w);
}

// ---------------------------------------------------------------------------
// Kernel 3: per (bt, 16-node tile): D[16x32] = A[16x16] @ W[16x32] + bias,
// via 4 chained V_WMMA_F32_16X16X4_F32 per 16-col output tile (2 tiles),
// then LayerNorm over the 32 channels with half-wave shuffle reductions.
// No divergence: grid exactly covers BT*(N/16) waves (EXEC all-1s for WMMA).
//
// ISA register layouts (wave32):
//   A 16x4 f32 : lane L (m=L&15, hi=L>>4): VGPR0=A[m][kb+2*hi], VGPR1=+1
//   B 4x16 f32 : lane L: VGPR0=W[kb+2*hi][n], VGPR1=W[kb+2*hi+1][n]
//   C/D 16x16  : VGPR i: lanes 0-15 -> row i, lanes 16-31 -> row i+8
// ---------------------------------------------------------------------------
__global__ void gemm_ln_kernel(const float* __restrict__ agg,
                               const float* __restrict__ Wf,    // [16][32]
                               const float* __restrict__ bias,  // [32]
                               const float* __restrict__ gamma, // [32]
                               const float* __restrict__ beta,  // [32]
                               float* __restrict__ out) {
  const int TILES = NNODES / 16;                       // 4096 tiles per bt
  int wid  = (blockIdx.x * blockDim.x + threadIdx.x) >> 5;
  int lane = threadIdx.x & 31;
  int bt   = wid / TILES;
  int tile = wid - bt * TILES;
  int n0   = tile * 16;
  int m    = lane & 15;
  int hi   = lane >> 4;                                // 0: lanes 0-15, 1: 16-31

  const float* arow = agg + (size_t)(n0 + m) * SLAB + bt * FIN;

  v8f c0 = {};
  v8f c1 = {};
#pragma unroll
  for (int s = 0; s < 4; ++s) {
    int kb = 4 * s + 2 * hi;
    v2f a;
    a.x = arow[kb + 0];
    a.y = arow[kb + 1];
    v2f b0, b1;
    b0.x = Wf[(kb + 0) * FOUT + m];
    b0.y = Wf[(kb + 1) * FOUT + m];
    b1.x = Wf[(kb + 0) * FOUT + 16 + m];
    b1.y = Wf[(kb + 1) * FOUT + 16 + m];
    c0 = __builtin_amdgcn_wmma_f32_16x16x4_f32(false, a, false, b0,
                                               (short)0, c0, false, false);
    c1 = __builtin_amdgcn_wmma_f32_16x16x4_f32(false, a, false, b1,
                                               (short)0, c1, false, false);
  }

  float bi0 = bias[m], bi1 = bias[16 + m];
  float g0 = gamma[m], g1 = gamma[16 + m];
  float be0 = beta[m], be1 = beta[16 + m];

#pragma unroll
  for (int i = 0; i < 8; ++i) {
    float v0 = c0[i] + bi0;                    // channel m       of row i+8*hi
    float v1 = c1[i] + bi1;                    // channel 16+m    of row i+8*hi
    float sum = v0 + v1;
    float sq  = v0 * v0 + v1 * v1;
    // reduce across the 16 lanes holding this row
#pragma unroll
    for (int off = 8; off >= 1; off >>= 1) {
      sum += __shfl_xor(sum, off, 16);
      sq  += __shfl_xor(sq,  off, 16);
    }
    float mu  = sum * (1.0f / 32.0f);
    float var = sq * (1.0f / 32.0f) - mu * mu;
    float inv = rsqrtf(var + 1e-3f);
    float o0 = (v0 - mu) * inv * g0 + be0;
    float o1 = (v1 - mu) * inv * g1 + be1;
    int node = n0 + i + 8 * hi;
    size_t base = ((size_t)bt * NNODES + node) * FOUT;
    out[base + m]      = o0;
    out[base + 16 + m] = o1;
  }
}

// ---------------------------------------------------------------------------
extern "C" void kernel_launch(void* const* d_in, const int* in_sizes, int n_in,
                              void* d_out, int out_size, void* d_ws, size_t ws_size,
                              hipStream_t stream) {
  const float* water    = (const float*)d_in[0];
  const float* stat     = (const float*)d_in[1];
  const int*   rows     = (const int*)  d_in[2];
  const int*   cols     = (const int*)  d_in[3];
  const float* vals     = (const float*)d_in[4];
  const float* Wf       = (const float*)d_in[5];
  const float* bf       = (const float*)d_in[6];
  const float* Wg1      = (const float*)d_in[7];
  const float* bg1      = (const float*)d_in[8];
  const float* Wg2      = (const float*)d_in[9];
  const float* bg2      = (const float*)d_in[10];
  const float* gamma    = (const float*)d_in[11];
  const float* beta     = (const float*)d_in[12];
  const float* min_gate = (const float*)d_in[13];
  float* out = (float*)d_out;
  int E = in_sizes[2];

  float* x   = (float*)d_ws;                                  // [N][BT][FIN]
  float* agg = x + (size_t)NNODES * SLAB;                     // [N][BT][FIN]

  // 1) gate + scale + zero agg: N*BT threads
  {
    int total = NNODES * BT;
    gate_scale_kernel<<<total / 256, 256, 0, stream>>>(
        water, stat, Wg1, bg1, Wg2, bg2, min_gate, x, agg);
  }
  // 2) SpMM scatter: one wave per edge
  {
    long long total = (long long)E * 32;
    int blocks = (int)((total + 255) / 256);
    spmm_kernel<<<blocks, 256, 0, stream>>>(rows, cols, vals, x, agg, E);
  }
  // 3) WMMA GEMM + LayerNorm: one wave per (bt, 16-node tile)
  {
    int waves  = BT * (NNODES / 16);     // 65536
    int blocks = waves * 32 / 256;       // 8192
    gemm_ln_kernel<<<blocks, 256, 0, stream>>>(agg, Wf, bf, gamma, beta, out);
  }
}